// MixtureOfExperts_27195732918639
// MI455X (gfx1250) — compile-verified
//
#include <hip/hip_runtime.h>

#define MOE_D 1024
#define MOE_E 8

typedef __attribute__((ext_vector_type(2))) float v2f;
typedef __attribute__((ext_vector_type(8))) float v8f;

// ---------------------------------------------------------------------------
// Router: logits -> softmax -> top-2 -> per-expert row lists + weights
// ---------------------------------------------------------------------------
__global__ __launch_bounds__(256) void moe_router_kernel(
    const float* __restrict__ x, const float* __restrict__ Wg,
    const float* __restrict__ bg, int T, int* __restrict__ cnt,
    int* __restrict__ rows0, int* __restrict__ rows1,
    float* __restrict__ w0, float* __restrict__ w1)
{
    __shared__ float4 wg_lds[MOE_E * (MOE_D / 4)];   // 32 KB of 320 KB WGP LDS
    const float4* wg4 = (const float4*)Wg;
    for (int i = threadIdx.x; i < MOE_E * MOE_D / 4; i += 256) wg_lds[i] = wg4[i];
    __syncthreads();

    int t = blockIdx.x * 256 + threadIdx.x;
    if (t >= T) return;

    float acc[MOE_E];
#pragma unroll
    for (int e = 0; e < MOE_E; ++e) acc[e] = bg[e];

    const float4* xv = (const float4*)(x + (size_t)t * MOE_D);
    for (int d4 = 0; d4 < MOE_D / 4; ++d4) {
        float4 xx = xv[d4];
#pragma unroll
        for (int e = 0; e < MOE_E; ++e) {
            float4 ww = wg_lds[e * (MOE_D / 4) + d4];
            acc[e] += xx.x * ww.x + xx.y * ww.y + xx.z * ww.z + xx.w * ww.w;
        }
    }

    // top-2 over logits (monotone with softmax probs); ties -> lowest index
    int e0 = 0;
#pragma unroll
    for (int e = 1; e < MOE_E; ++e) if (acc[e] > acc[e0]) e0 = e;
    int e1 = (e0 == 0) ? 1 : 0;
#pragma unroll
    for (int e = 0; e < MOE_E; ++e) if (e != e0 && acc[e] > acc[e1]) e1 = e;

    float m = acc[e0];
    float sum = 0.f;
#pragma unroll
    for (int e = 0; e < MOE_E; ++e) sum += __expf(acc[e] - m);
    float inv = 1.0f / sum;
    float p0 = __expf(acc[e0] - m) * inv;
    float p1 = __expf(acc[e1] - m) * inv;
    w0[t] = p0;
    w1[t] = p1;

    int pos0 = atomicAdd(&cnt[e0], 1);
    rows0[e0 * T + pos0] = t;
    int pos1 = atomicAdd(&cnt[MOE_E + e1], 1);
    rows1[e1 * T + pos1] = t;
}

// ---------------------------------------------------------------------------
// Gathered expert GEMM using V_WMMA_F32_16X16X4_F32.
// Wave tile: 32 rows (2x 16x16 M-tiles) x 64 cols (4x 16x16 N-tiles).
// Block: 8 waves -> 32 rows x 512 cols.  Grid: (experts, row-tiles, N-halves).
// out[t][n] (=/+=) p_t * (sum_k x[t][k] * W[e][n][k] + b[e][n])
// ---------------------------------------------------------------------------
__global__ __launch_bounds__(256) void moe_expert_gemm_kernel(
    const float* __restrict__ x, const float* __restrict__ W,
    const float* __restrict__ bexp, const int* __restrict__ cnt,
    const int* __restrict__ rows, const float* __restrict__ wt,
    float* __restrict__ out, int T, int accumulate)
{
    const int e = blockIdx.x;
    const int n_rows = cnt[e];
    const int rbase = blockIdx.y * 32;
    if (rbase >= n_rows) return;               // uniform early-exit (EXEC intact)

    const int lane  = threadIdx.x & 31;
    const int wave  = threadIdx.x >> 5;
    const int mrow  = lane & 15;               // M (A) or N (B/C/D) position
    const int khalf = lane >> 4;               // K pair select / M-half select
    const int ncol0 = blockIdx.z * 512 + wave * 64;

    const int* rl = rows + (size_t)e * T + rbase;
    // padded rows read x row 0: finite garbage, never stored
    int tok0 = (rbase + mrow < n_rows) ? rl[mrow] : 0;
    int tok1 = (rbase + 16 + mrow < n_rows) ? rl[16 + mrow] : 0;
    const float* xr0 = x + (size_t)tok0 * MOE_D + khalf * 2;
    const float* xr1 = x + (size_t)tok1 * MOE_D + khalf * 2;
    const float* wb  = W + (size_t)e * MOE_D * MOE_D
                         + (size_t)(ncol0 + mrow) * MOE_D + khalf * 2;

    const v8f vzero = {0.f, 0.f, 0.f, 0.f, 0.f, 0.f, 0.f, 0.f};
    v8f c[2][4];
#pragma unroll
    for (int g = 0; g < 2; ++g)
#pragma unroll
        for (int s = 0; s < 4; ++s) c[g][s] = vzero;

#pragma unroll 2
    for (int k0 = 0; k0 < MOE_D; k0 += 4) {
        // A frags: lane m<16 holds A[m][k0+0..1], lane m+16 holds A[m][k0+2..3]
        v2f a0 = *(const v2f*)(xr0 + k0);
        v2f a1 = *(const v2f*)(xr1 + k0);
#pragma unroll
        for (int s = 0; s < 4; ++s) {
            // B frag: lane n<16 holds B[k0+0..1][n], lane n+16 holds B[k0+2..3][n]
            v2f bb = *(const v2f*)(wb + (size_t)s * 16 * MOE_D + k0);
            c[0][s] = __builtin_amdgcn_wmma_f32_16x16x4_f32(
                false, a0, false, bb, (short)0, c[0][s], false, false);
            c[1][s] = __builtin_amdgcn_wmma_f32_16x16x4_f32(
                false, a1, false, bb, (short)0, c[1][s], false, false);
        }
    }

    // Epilogue. C/D layout: VGPR v, lanes 0-15 -> (M=v, N=lane),
    //                       lanes 16-31 -> (M=v+8, N=lane-16)
    float bias[4];
#pragma unroll
    for (int s = 0; s < 4; ++s)
        bias[s] = bexp[e * MOE_D + ncol0 + s * 16 + mrow];

#pragma unroll
    for (int g = 0; g < 2; ++g) {
#pragma unroll
        for (int v = 0; v < 8; ++v) {
            int m = g * 16 + v + khalf * 8;
            if (rbase + m < n_rows) {
                int t = rl[m];
                float p = wt[t];
                float* orow = out + (size_t)t * MOE_D + ncol0 + mrow;
#pragma unroll
                for (int s = 0; s < 4; ++s) {
                    float val = p * (c[g][s][v] + bias[s]);
                    if (accumulate) orow[s * 16] += val;   // unique writer
                    else            orow[s * 16]  = val;
                }
            }
        }
    }
}

// ---------------------------------------------------------------------------
extern "C" void kernel_launch(void* const* d_in, const int* in_sizes, int n_in,
                              void* d_out, int out_size, void* d_ws, size_t ws_size,
                              hipStream_t stream) {
    const float* x  = (const float*)d_in[0];   // [4,2048,1024]
    const float* We = (const float*)d_in[1];   // [8,1024,1024]
    const float* be = (const float*)d_in[2];   // [8,1024]
    const float* Wg = (const float*)d_in[3];   // [8,1024]
    const float* bg = (const float*)d_in[4];   // [8]
    float* out = (float*)d_out;
    const int T = in_sizes[0] / MOE_D;         // 8192

    // workspace layout (~590 KB)
    char* ws  = (char*)d_ws;
    int* cnt   = (int*)ws;                     // 16 counters (slot0: 0..7, slot1: 8..15)
    int* rows0 = (int*)(ws + 256);             // [E][T]
    int* rows1 = rows0 + MOE_E * T;            // [E][T]
    float* w0  = (float*)(rows1 + MOE_E * T);  // [T]
    float* w1  = w0 + T;                       // [T]

    hipMemsetAsync(cnt, 0, 16 * sizeof(int), stream);

    moe_router_kernel<<<(T + 255) / 256, 256, 0, stream>>>(
        x, Wg, bg, T, cnt, rows0, rows1, w0, w1);

    dim3 grid(MOE_E, (T + 31) / 32, 2);
    // slot 0: plain stores (every token appears exactly once -> full init)
    moe_expert_gemm_kernel<<<grid, 256, 0, stream>>>(
        x, We, be, cnt, rows0, w0, out, T, 0);
    // slot 1: accumulate (ordered after slot 0 on the stream)
    moe_expert_gemm_kernel<<<grid, 256, 0, stream>>>(
        x, We, be, cnt + MOE_E, rows1, w1, out, T, 1);
}